// TwinGCN_43662637531790
// MI455X (gfx1250) — compile-verified
//
#include <hip/hip_runtime.h>

typedef __attribute__((ext_vector_type(2))) float v2f;
typedef __attribute__((ext_vector_type(8))) float v8f;

// ---------------- degree / normalization ----------------

__global__ __launch_bounds__(256) void k_init_deg(float* deg, int n) {
  int i = blockIdx.x * blockDim.x + threadIdx.x;
  if (i < n) deg[i] = 1.0f;  // self-loop
}

__global__ __launch_bounds__(256) void k_accum_deg(const int* __restrict__ dst,
                                                   float* deg, int e) {
  int i = blockIdx.x * blockDim.x + threadIdx.x;
  if (i < e) atomicAdd(&deg[dst[i]], 1.0f);
}

__global__ __launch_bounds__(256) void k_make_dinv(float* deg, int n) {
  int i = blockIdx.x * blockDim.x + threadIdx.x;
  if (i < n) { float d = deg[i]; deg[i] = (d > 0.0f) ? rsqrtf(d) : 0.0f; }
}

// ---- f32 WMMA GEMM: C[MxNc] = A[MxK] @ B[KxNc] (+bias, +relu) --------------
// One wave computes a 64x16 output block: 4 stacked 16x16 tiles sharing each
// B fragment load -> 4 independent v_wmma_f32_16x16x4_f32 per k-step.
// A 16x4 frag: lanes 0-15 hold K={0,1}, lanes 16-31 hold K={2,3}; B mirrored.
// C/D frag: vgpr i holds row (i + 8*(lane>=16)), col = lane&15.

template <bool COLCHK>
__global__ __launch_bounds__(256) void k_wmma_gemm(
    const float* __restrict__ A, const float* __restrict__ B,
    const float* __restrict__ bias, float* __restrict__ C,
    int M, int K, int Nc, int tilesN, int doRelu)
{
  int wave   = blockIdx.x * (blockDim.x >> 5) + (threadIdx.x >> 5);
  int tilesM = (M + 63) >> 6;
  if (wave >= tilesM * tilesN) return;           // whole-wave exit only
  int tm = wave / tilesN, tn = wave % tilesN;
  int lane = threadIdx.x & 31;
  int half = lane >> 4;
  int r    = lane & 15;

  int rowBase = tm * 64 + r;
  int mr      = M - 1;
  const float* __restrict__ A0 = A + (size_t)min(rowBase,      mr) * K;
  const float* __restrict__ A1 = A + (size_t)min(rowBase + 16, mr) * K;
  const float* __restrict__ A2 = A + (size_t)min(rowBase + 32, mr) * K;
  const float* __restrict__ A3 = A + (size_t)min(rowBase + 48, mr) * K;

  int  bcol = tn * 16 + r;
  bool bOK  = !COLCHK || (bcol < Nc);
  int  lcol = bOK ? bcol : 0;                    // clamped load column

  v8f acc0 = {}, acc1 = {}, acc2 = {}, acc3 = {};
  for (int k = 0; k < K; k += 4) {
    int ka = k + half * 2;
    v2f b;
    b.x = B[(size_t)ka * Nc + lcol];
    b.y = B[(size_t)(ka + 1) * Nc + lcol];
    if (COLCHK) {                                 // select, not branch
      b.x = bOK ? b.x : 0.0f;
      b.y = bOK ? b.y : 0.0f;
    }
    v2f a0 = { A0[ka], A0[ka + 1] };
    v2f a1 = { A1[ka], A1[ka + 1] };
    v2f a2 = { A2[ka], A2[ka + 1] };
    v2f a3 = { A3[ka], A3[ka + 1] };
    acc0 = __builtin_amdgcn_wmma_f32_16x16x4_f32(false, a0, false, b, (short)0, acc0, false, false);
    acc1 = __builtin_amdgcn_wmma_f32_16x16x4_f32(false, a1, false, b, (short)0, acc1, false, false);
    acc2 = __builtin_amdgcn_wmma_f32_16x16x4_f32(false, a2, false, b, (short)0, acc2, false, false);
    acc3 = __builtin_amdgcn_wmma_f32_16x16x4_f32(false, a3, false, b, (short)0, acc3, false, false);
  }

  if (bOK) {
    float bv = bias ? bias[bcol] : 0.0f;
    v8f accs[4] = { acc0, acc1, acc2, acc3 };
#pragma unroll
    for (int t = 0; t < 4; ++t) {
#pragma unroll
      for (int i = 0; i < 8; ++i) {
        int crow = tm * 64 + t * 16 + half * 8 + i;
        if (crow < M) {
          float v = accs[t][i] + bv;
          if (doRelu) v = fmaxf(v, 0.0f);
          C[(size_t)crow * Nc + bcol] = v;
        }
      }
    }
  }
}

// ---------------- sparse propagate ----------------

// P[n,:] = dinv[n]^2 * T[n,:]   (self-loop term, also zero-bases P)
__global__ __launch_bounds__(256) void k_self_init(const float* __restrict__ dinv,
                                                   const float* __restrict__ T,
                                                   float* __restrict__ P, int n64) {
  int i = blockIdx.x * blockDim.x + threadIdx.x;
  if (i < n64) { float w = dinv[i >> 6]; P[i] = w * w * T[i]; }
}

// one wave per edge: P[dst,:] += dinv[src]*dinv[dst] * T[src,:]
// lane handles features {2*lane, 2*lane+1}: one b64 load + two f32 atomics.
__global__ __launch_bounds__(256) void k_edge_scatter(
    const int* __restrict__ src, const int* __restrict__ dst,
    const float* __restrict__ dinv, const float* __restrict__ T,
    float* __restrict__ P, int e)
{
  int eid = blockIdx.x * 8 + (threadIdx.x >> 5);
  if (eid >= e) return;
  int s = src[eid], d = dst[eid];
  float w = dinv[s] * dinv[d];
  int lane = threadIdx.x & 31;
  const float2* __restrict__ hs = (const float2*)(T + (size_t)s * 64);
  float2 v = hs[lane];
  float* __restrict__ pd = P + (size_t)d * 64 + lane * 2;
  atomicAdd(pd,     v.x * w);
  atomicAdd(pd + 1, v.y * w);
}

__global__ __launch_bounds__(256) void k_bias_relu(const float* __restrict__ P,
                                                   const float* __restrict__ b,
                                                   float* __restrict__ H, int n64) {
  int i = blockIdx.x * blockDim.x + threadIdx.x;
  if (i < n64) H[i] = fmaxf(P[i] + b[i & 63], 0.0f);
}

// ---------------- summarize ----------------

// Bqk[c,d] = (1/TEMP) * sum_e Wq[d,e]*Wk[c,e]   (i.e. (Wq Wk^T)^T / TEMP)
// so that (h @ Bqk)[n,d] = (Mqk h_n)_d and score = dot(h_n, h @ Bqk).
__global__ __launch_bounds__(256) void k_wqk(const float* __restrict__ Wq,
                                             const float* __restrict__ Wk,
                                             float* __restrict__ Bqk, float scale) {
  int i = blockIdx.x * blockDim.x + threadIdx.x;   // 4096 threads
  int c = i >> 6, d = i & 63;
  float s = 0.0f;
#pragma unroll 8
  for (int e = 0; e < 64; ++e) s += Wq[d * 64 + e] * Wk[c * 64 + e];
  Bqk[i] = s * scale;
}

// S[n] = dot(H[n,:], G[n,:])
__global__ __launch_bounds__(256) void k_rowdot(const float* __restrict__ Hrow,
                                                const float* __restrict__ G,
                                                float* __restrict__ S, int n) {
  int i = blockIdx.x * blockDim.x + threadIdx.x;
  if (i >= n) return;
  const float* a = Hrow + (size_t)i * 64;
  const float* g = G + (size_t)i * 64;
  float s = 0.0f;
#pragma unroll 8
  for (int f = 0; f < 64; ++f) s += a[f] * g[f];
  S[i] = s;
}

// softmax over L=4 layers + weighted sum of Hs -> out
__global__ __launch_bounds__(256) void k_softmax_combine(
    const float* __restrict__ S, const float* __restrict__ Hs,
    float* __restrict__ out, int n, int n64)
{
  int i = blockIdx.x * blockDim.x + threadIdx.x;
  if (i >= n64) return;
  int node = i >> 6;
  float s0 = S[node];
  float s1 = S[(size_t)n + node];
  float s2 = S[2 * (size_t)n + node];
  float s3 = S[3 * (size_t)n + node];
  float m = fmaxf(fmaxf(s0, s1), fmaxf(s2, s3));
  float e0 = __expf(s0 - m), e1 = __expf(s1 - m), e2 = __expf(s2 - m), e3 = __expf(s3 - m);
  float inv = 1.0f / (e0 + e1 + e2 + e3);
  size_t st = (size_t)n64;
  out[i] = (e0 * Hs[i] + e1 * Hs[st + i] + e2 * Hs[2 * st + i] + e3 * Hs[3 * st + i]) * inv;
}

// ---------------- host orchestration ----------------

extern "C" void kernel_launch(void* const* d_in, const int* in_sizes, int n_in,
                              void* d_out, int out_size, void* d_ws, size_t ws_size,
                              hipStream_t stream) {
  const float* x    = (const float*)d_in[0];
  const int*   ei   = (const int*)d_in[1];   // [2, E] int32
  const float* W0   = (const float*)d_in[2]; // [128, 64]
  const float* b0   = (const float*)d_in[3]; // [64]
  const float* Ws   = (const float*)d_in[4]; // [3, 64, 64]
  const float* bs   = (const float*)d_in[5]; // [3, 64]
  const float* Wq   = (const float*)d_in[6]; // [64, 64]
  const float* Wk   = (const float*)d_in[7]; // [64, 64]
  const float* Wout = (const float*)d_in[8]; // [64, 40]
  const float* bout = (const float*)d_in[9]; // [40]
  (void)n_in; (void)ws_size;

  const int F = 128, H = 64, L = 4;
  const int N = in_sizes[0] / F;
  const int E = in_sizes[1] / 2;
  const int C = out_size / N;
  const int* src = ei;
  const int* dst = ei + E;

  // workspace layout (floats)
  float* f    = (float*)d_ws;
  float* dinv = f;                          // N      (deg -> dinv in place)
  float* T    = dinv + N;                   // N*64   gemm output
  float* P    = T + (size_t)N * H;          // N*64   propagate accum / hsum
  float* Hs   = P + (size_t)N * H;          // L*N*64 layer outputs
  float* S    = Hs + (size_t)L * N * H;     // L*N    attention scores
  float* Bqk  = S + (size_t)L * N;          // 64*64  (Wq Wk^T)^T / TEMP

  const int n64 = N * H;
  dim3 blk(256);
  auto cdiv = [](int a, int b) { return (a + b - 1) / b; };

  // GCN normalization: deg -> dinv
  k_init_deg <<<cdiv(N, 256), blk, 0, stream>>>(dinv, N);
  k_accum_deg<<<cdiv(E, 256), blk, 0, stream>>>(dst, dinv, E);
  k_make_dinv<<<cdiv(N, 256), blk, 0, stream>>>(dinv, N);

  // combined attention matrix (tiny, overlaps with deg work)
  k_wqk<<<16, blk, 0, stream>>>(Wq, Wk, Bqk, 10.0f /* 1/TEMP */);

  const int tilesM = cdiv(N, 64);           // 64 rows per wave
  const int tilesH = cdiv(H, 16);
  const int wavesH = tilesM * tilesH;

  // Twin branches are identical (same init, same weights, eval mode) -> compute once.
  for (int l = 0; l < L; ++l) {
    const float* hin = (l == 0) ? x : (Hs + (size_t)(l - 1) * n64);
    const float* W   = (l == 0) ? W0 : (Ws + (size_t)(l - 1) * H * H);
    const float* b   = (l == 0) ? b0 : (bs + (size_t)(l - 1) * H);
    const int K      = (l == 0) ? F : H;
    k_wmma_gemm<false><<<cdiv(wavesH, 8), blk, 0, stream>>>(hin, W, nullptr, T, N, K, H, tilesH, 0);
    k_self_init   <<<cdiv(n64, 256), blk, 0, stream>>>(dinv, T, P, n64);
    k_edge_scatter<<<cdiv(E, 8),     blk, 0, stream>>>(src, dst, dinv, T, P, E);
    k_bias_relu   <<<cdiv(n64, 256), blk, 0, stream>>>(P, b, Hs + (size_t)l * n64, n64);
  }

  // scores[l,n] = h_n^T (WqWk^T/TEMP) h_n  via one GEMM + rowdot per layer
  for (int l = 0; l < L; ++l) {
    const float* hl = Hs + (size_t)l * n64;
    k_wmma_gemm<false><<<cdiv(wavesH, 8), blk, 0, stream>>>(hl, Bqk, nullptr, T, N, H, H, tilesH, 0);
    k_rowdot<<<cdiv(N, 256), blk, 0, stream>>>(hl, T, S + (size_t)l * N, N);
  }

  // softmax over layers + weighted layer sum -> P (hsum)
  k_softmax_combine<<<cdiv(n64, 256), blk, 0, stream>>>(S, Hs, P, N, n64);

  // out = hsum @ Wout + bout
  const int tilesC = cdiv(C, 16);
  k_wmma_gemm<true><<<cdiv(tilesM * tilesC, 8), blk, 0, stream>>>(P, Wout, bout, (float*)d_out,
                                                                  N, H, C, tilesC, 0);
}